// LastAggregator_14955076125084
// MI455X (gfx1250) — compile-verified
//
#include <hip/hip_runtime.h>
#include <stdint.h>

#define D_DIM 128  // feature dim per reference

// ---------------------------------------------------------------------------
// Phase 1: init per-node packed keys to 0 (0 == "empty node" sentinel).
// key layout: [63:32] = monotone-mapped float bits of t, [31:0] = pos+1.
// ---------------------------------------------------------------------------
__global__ __launch_bounds__(256) void la_init_keys(
    unsigned long long* __restrict__ keys, int n) {
  int i = blockIdx.x * blockDim.x + threadIdx.x;
  if (i < n) keys[i] = 0ULL;
}

// ---------------------------------------------------------------------------
// Phase 2: scatter-argmax, 4 events per thread.
// One B128 load of t[4], one B128 load of index[4], then 4 non-returning
// global_atomic_max_u64 (fire-and-forget, STOREcnt-tracked, resolved in L2).
// Monotone float->uint map is order-preserving for all floats; for t in [0,1)
// it reduces to tb | 0x8000_0000. Low 32 bits carry pos+1, so max-key yields
// (max t, largest position among ties) — exactly the reference tie-break.
// ---------------------------------------------------------------------------
__device__ __forceinline__ unsigned long long la_pack_key(float tv, int e) {
  unsigned int tb = __float_as_uint(tv);
  unsigned int mt = (tb & 0x80000000u) ? ~tb : (tb | 0x80000000u);
  return ((unsigned long long)mt << 32) |
         (unsigned long long)(unsigned int)(e + 1);
}

__global__ __launch_bounds__(256) void la_scatter_max4(
    const float* __restrict__ t,
    const int* __restrict__ index,
    unsigned long long* __restrict__ keys,
    int num_events) {
  int base = (blockIdx.x * blockDim.x + threadIdx.x) * 4;
  if (base + 3 < num_events) {
    const float4 tv = *(const float4*)(t + base);
    const int4   iv = *(const int4*)(index + base);
    atomicMax(&keys[iv.x], la_pack_key(tv.x, base + 0));
    atomicMax(&keys[iv.y], la_pack_key(tv.y, base + 1));
    atomicMax(&keys[iv.z], la_pack_key(tv.z, base + 2));
    atomicMax(&keys[iv.w], la_pack_key(tv.w, base + 3));
  } else {
    for (int e = base; e < num_events; ++e) {
      atomicMax(&keys[index[e]], la_pack_key(t[e], e));
    }
  }
}

// ---------------------------------------------------------------------------
// Phase 3: gather winning rows. One wave32 per node; lane l moves 16 bytes
// [16l, 16l+16) of the 512B row through the CDNA5 async path:
//   global_load_async_to_lds_b128   (memory -> LDS, ASYNCcnt)
//   s_wait_asynccnt 0
//   global_store_async_from_lds_b128 (LDS -> memory, ASYNCcnt)
// so the row never round-trips through VGPRs. Each wave owns a private 512B
// LDS stage slice (8 waves/block -> 4KB LDS), no barriers needed; row-copy
// latency is hidden by ~100K independent waves.
// Empty nodes (key==0, ~e^-10 of nodes) store a zero float4 directly.
// ---------------------------------------------------------------------------
__global__ __launch_bounds__(256) void la_gather_async(
    const float* __restrict__ msg,
    const unsigned long long* __restrict__ keys,
    float* __restrict__ out, int n) {
  __shared__ __align__(16) unsigned char stage[8 * 512];

  const int lane = threadIdx.x & 31;
  const int widx = threadIdx.x >> 5;          // wave index within block
  const int node = blockIdx.x * 8 + widx;
  if (node >= n) return;                      // wave-uniform exit

  const unsigned long long key = keys[node];
  float4* dst = (float4*)(out + (size_t)node * D_DIM) + lane;

  if (key == 0ULL) {                          // empty segment -> zeros
    *dst = make_float4(0.f, 0.f, 0.f, 0.f);
    return;
  }

  const size_t pos = (size_t)((unsigned int)key - 1u);
  const unsigned long long src =
      (unsigned long long)(const char*)(msg + pos * D_DIM) +
      (unsigned long long)(lane * 16);
  // LDS byte offset of this lane's 16B slot (generic LDS addr low 32 bits
  // == workgroup-relative LDS offset per CDNA5 aperture rules).
  const unsigned int lds_off =
      (unsigned int)(size_t)(&stage[widx * 512 + lane * 16]);
  const unsigned long long dsta = (unsigned long long)(size_t)dst;

  asm volatile("global_load_async_to_lds_b128 %0, %1, off"
               :: "v"(lds_off), "v"(src)
               : "memory");
  asm volatile("s_wait_asynccnt 0" ::: "memory");
  asm volatile("global_store_async_from_lds_b128 %0, %1, off"
               :: "v"(dsta), "v"(lds_off)
               : "memory");
  asm volatile("s_wait_asynccnt 0" ::: "memory");
}

// ---------------------------------------------------------------------------
// Harness entry. Inputs (setup_inputs order): msg[E*D] f32, index[E] int,
// t[E] f32, dim_size (scalar, unused — N derived from out_size).
// Workspace: N * 8 bytes of packed keys (re-initialized every call, so the
// launch is deterministic and safe under 0xAA poisoning).
// ---------------------------------------------------------------------------
extern "C" void kernel_launch(void* const* d_in, const int* in_sizes, int n_in,
                              void* d_out, int out_size, void* d_ws, size_t ws_size,
                              hipStream_t stream) {
  const float* msg   = (const float*)d_in[0];
  const int*   index = (const int*)d_in[1];
  const float* t     = (const float*)d_in[2];
  (void)n_in; (void)ws_size;

  const int E_ = in_sizes[2];          // number of events
  const int N_ = out_size / D_DIM;     // number of nodes

  unsigned long long* keys = (unsigned long long*)d_ws;

  la_init_keys<<<(N_ + 255) / 256, 256, 0, stream>>>(keys, N_);

  const int ev_threads = (E_ + 3) / 4;
  la_scatter_max4<<<(ev_threads + 255) / 256, 256, 0, stream>>>(t, index, keys, E_);

  la_gather_async<<<(N_ + 7) / 8, 256, 0, stream>>>(msg, keys, (float*)d_out, N_);
}